// Net_42322607735228
// MI455X (gfx1250) — compile-verified
//
#include <hip/hip_runtime.h>

typedef float v2f __attribute__((ext_vector_type(2)));
typedef float v8f __attribute__((ext_vector_type(8)));

#define WAVES_PER_BLOCK 4
#define STAGE_STRIDE 66               // 16-row staging tile, odd-ish dword stride vs 64 banks
#define STAGE_FLOATS (16 * STAGE_STRIDE)   // 1056 floats per buffer

// ---- shared memory layout (float offsets) ----
#define OFF_L1F 0        // W1 frags: 32 ksteps * 4 ntiles * 64 = 8192
#define OFF_L2F 8192     // W2 frags: 16 * 2 * 64 = 2048
#define OFF_L3F 10240    // W3 frags:  8 * 1 * 64 = 512  (N padded 3->16)
#define OFF_D1F 10752    // V1 frags:  1 * 2 * 64 = 128  (K padded 3->4)
#define OFF_D2F 10880    // V2 frags:  8 * 4 * 64 = 2048
#define OFF_D3F 12928    // V3 frags: 16 * 8 * 64 = 8192
#define OFF_B1  21120
#define OFF_B2  21184
#define OFF_B3  21216
#define OFF_C1  21220
#define OFF_C2  21252
#define OFF_C3  21316
#define OFF_EW  21444    // 3x22
#define OFF_EB  21510
#define OFF_STAGE 21520
#define SMEM_FLOATS (OFF_STAGE + WAVES_PER_BLOCK * 2 * STAGE_FLOATS)  // 29968 floats ~ 117 KB

__device__ __forceinline__ v8f v8zero() {
  v8f z = {0.f, 0.f, 0.f, 0.f, 0.f, 0.f, 0.f, 0.f};
  return z;
}

__device__ __forceinline__ v8f wmma_f32(v2f a, v2f b, v8f c) {
  // V_WMMA_F32_16X16X4_F32: D = A(16x4,f32) x B(4x16,f32) + C(16x16,f32)
  return __builtin_amdgcn_wmma_f32_16x16x4_f32(false, a, false, b, (short)0, c, false, false);
}

// Pre-swizzle a row-major weight W[Nout][K] (as W.T is the GEMM B-matrix) into
// fragment-major LDS: frag[(kstep*ntiles + nt)*64 + lane*2 + j] = B[k][n]
// with n = nt*16 + (lane&15), k = kstep*4 + (lane>>4)*2 + j. Zero-pad OOB.
__device__ __forceinline__ void build_frags(float* dst, const float* __restrict__ W,
                                            int Nout, int K, int ksteps, int ntiles,
                                            int tid, int nthr) {
  int total = ksteps * ntiles * 64;
  for (int idx = tid; idx < total; idx += nthr) {
    int kstep = idx / (ntiles * 64);
    int rem   = idx - kstep * ntiles * 64;
    int nt    = rem >> 6;
    int j     = rem & 63;
    int lane  = j >> 1;
    int which = j & 1;
    int nn = nt * 16 + (lane & 15);
    int kk = kstep * 4 + ((lane >> 4) << 1) + which;
    dst[idx] = (nn < Nout && kk < K) ? W[nn * K + kk] : 0.0f;
  }
}

__device__ __forceinline__ void copy_f(float* dst, const float* __restrict__ src,
                                       int cnt, int tid, int nthr) {
  for (int i = tid; i < cnt; i += nthr) dst[i] = src[i];
}

// Dual-stream GEMM: value and tangent A-tiles (row-major, stride astride)
// against shared B-fragments in LDS. Accumulates NTILES 16x16 fp32 tiles each.
template <int KSTEPS, int NTILES>
__device__ __forceinline__ void dual_gemm(const float* Aval, const float* Atan, int astride,
                                          const float* Bfrag, int lane,
                                          v8f* vacc, v8f* tacc) {
  const int abase = (lane & 15) * astride + ((lane >> 4) << 1);
  const int bbase = lane << 1;
#pragma unroll
  for (int k = 0; k < KSTEPS; ++k) {
    v2f av = *(const v2f*)(Aval + abase + k * 4);
    v2f at = *(const v2f*)(Atan + abase + k * 4);
#pragma unroll
    for (int nt = 0; nt < NTILES; ++nt) {
      v2f b = *(const v2f*)(Bfrag + (k * NTILES + nt) * 64 + bbase);
      vacc[nt] = wmma_f32(av, b, vacc[nt]);
      tacc[nt] = wmma_f32(at, b, tacc[nt]);
    }
  }
}

// bias + sigmoid (value) and h*(1-h)*du (tangent), store both 16-row tiles to staging.
template <int NTILES>
__device__ __forceinline__ void activate_stage(const v8f* vacc, const v8f* tacc,
                                               const float* bias,
                                               float* valbuf, float* tanbuf, int lane) {
  const int c = lane & 15, half = lane >> 4;
#pragma unroll
  for (int nt = 0; nt < NTILES; ++nt) {
    int col = nt * 16 + c;
    float b = bias[col];
#pragma unroll
    for (int r = 0; r < 8; ++r) {
      int row = r + half * 8;
      float u = vacc[nt][r] + b;
      float h = 1.0f / (1.0f + __expf(-u));
      float d = h * (1.0f - h) * tacc[nt][r];
      valbuf[row * STAGE_STRIDE + col] = h;
      tanbuf[row * STAGE_STRIDE + col] = d;
    }
  }
}

__global__ void __launch_bounds__(32 * WAVES_PER_BLOCK, 1)
sindy_ae_kernel(const float* __restrict__ x,  const float* __restrict__ dx,
                const float* __restrict__ W1, const float* __restrict__ b1,
                const float* __restrict__ W2, const float* __restrict__ b2,
                const float* __restrict__ W3, const float* __restrict__ b3,
                const float* __restrict__ V1, const float* __restrict__ c1,
                const float* __restrict__ V2, const float* __restrict__ c2,
                const float* __restrict__ V3, const float* __restrict__ c3,
                const float* __restrict__ Ew, const float* __restrict__ Eb,
                float* __restrict__ out, int nsamples) {
  extern __shared__ float smem[];
  const int tid = threadIdx.x;
  const int nthr = blockDim.x;

  // ---- stage weights as WMMA B-fragments + biases into LDS (once per block) ----
  build_frags(smem + OFF_L1F, W1, 64, 128, 32, 4, tid, nthr);
  build_frags(smem + OFF_L2F, W2, 32,  64, 16, 2, tid, nthr);
  build_frags(smem + OFF_L3F, W3,  3,  32,  8, 1, tid, nthr);
  build_frags(smem + OFF_D1F, V1, 32,   3,  1, 2, tid, nthr);
  build_frags(smem + OFF_D2F, V2, 64,  32,  8, 4, tid, nthr);
  build_frags(smem + OFF_D3F, V3, 128, 64, 16, 8, tid, nthr);
  copy_f(smem + OFF_B1, b1,  64, tid, nthr);
  copy_f(smem + OFF_B2, b2,  32, tid, nthr);
  copy_f(smem + OFF_B3, b3,   3, tid, nthr);
  copy_f(smem + OFF_C1, c1,  32, tid, nthr);
  copy_f(smem + OFF_C2, c2,  64, tid, nthr);
  copy_f(smem + OFF_C3, c3, 128, tid, nthr);
  copy_f(smem + OFF_EW, Ew,  66, tid, nthr);
  copy_f(smem + OFF_EB, Eb,   3, tid, nthr);
  __syncthreads();

  const int wave = tid >> 5;
  const int lane = tid & 31;
  const int tile = blockIdx.x * WAVES_PER_BLOCK + wave;
  if (tile * 16 >= nsamples) return;   // wave-uniform: EXEC stays all-ones for WMMA
  const int row0 = tile * 16;

  float* valbuf = smem + OFF_STAGE + wave * 2 * STAGE_FLOATS;
  float* tanbuf = valbuf + STAGE_FLOATS;

  const int c = lane & 15, half = lane >> 4;
  const size_t N = (size_t)nsamples;
  float* out_z   = out;
  float* out_dz  = out + 3 * N;
  float* out_dzb = out + 6 * N;
  float* out_xb  = out + 9 * N;
  float* out_dxb = out + 137 * N;

  // ---- encoder layer 1: 128 -> 64 (A streamed straight from HBM) ----
  v8f va[8], ta[8];
#pragma unroll
  for (int i = 0; i < 4; ++i) { va[i] = v8zero(); ta[i] = v8zero(); }
  dual_gemm<32, 4>(x + (size_t)row0 * 128, dx + (size_t)row0 * 128, 128,
                   smem + OFF_L1F, lane, va, ta);
  activate_stage<4>(va, ta, smem + OFF_B1, valbuf, tanbuf, lane);

  // ---- encoder layer 2: 64 -> 32 ----
#pragma unroll
  for (int i = 0; i < 2; ++i) { va[i] = v8zero(); ta[i] = v8zero(); }
  dual_gemm<16, 2>(valbuf, tanbuf, STAGE_STRIDE, smem + OFF_L2F, lane, va, ta);
  activate_stage<2>(va, ta, smem + OFF_B2, valbuf, tanbuf, lane);

  // ---- encoder layer 3 (linear): 32 -> 3 (N padded to 16, pad cols are 0) ----
  va[0] = v8zero(); ta[0] = v8zero();
  dual_gemm<8, 1>(valbuf, tanbuf, STAGE_STRIDE, smem + OFF_L3F, lane, va, ta);
  {
    float bz = (c < 3) ? smem[OFF_B3 + c] : 0.0f;
#pragma unroll
    for (int r = 0; r < 8; ++r) {
      int row = r + half * 8;
      float zv  = va[0][r] + bz;   // pad cols: 0 + 0
      float dzv = ta[0][r];
      valbuf[row * STAGE_STRIDE + c] = zv;
      tanbuf[row * STAGE_STRIDE + c] = dzv;
      if (c < 3) {
        out_z [(size_t)(row0 + row) * 3 + c] = zv;
        out_dz[(size_t)(row0 + row) * 3 + c] = dzv;
      }
    }
  }

  // ---- SINDy library + Ew/Eb: per-sample on lanes 0..15 (LDS is in-order per wave) ----
  if (lane < 16) {
    int row = lane;
    float z0 = valbuf[row * STAGE_STRIDE + 0];
    float z1 = valbuf[row * STAGE_STRIDE + 1];
    float z2 = valbuf[row * STAGE_STRIDE + 2];
    float th[22];
    th[0] = 1.0f; th[1] = 1.0f; th[2] = 1.0f;
    th[3] = z0;   th[4] = z1;   th[5] = z2;
    th[6]  = z0 * z0; th[7]  = z0 * z1; th[8]  = z0 * z2;
    th[9]  = z1 * z1; th[10] = z1 * z2; th[11] = z2 * z2;
    th[12] = th[6] * z0;  th[13] = th[6] * z1;  th[14] = th[6] * z2;
    th[15] = th[7] * z1;  th[16] = th[7] * z2;  th[17] = th[8] * z2;
    th[18] = th[9] * z1;  th[19] = th[9] * z2;  th[20] = th[10] * z2;
    th[21] = th[11] * z2;
#pragma unroll
    for (int r = 0; r < 3; ++r) {
      float acc = smem[OFF_EB + r];
#pragma unroll
      for (int t = 0; t < 22; ++t) acc += smem[OFF_EW + r * 22 + t] * th[t];
      out_dzb[(size_t)(row0 + row) * 3 + r] = acc;
      tanbuf[row * STAGE_STRIDE + r] = acc;  // tangent input for decoder (col 3 stays 0)
    }
  }

  // ---- decoder layer 1: 3 -> 32 (K padded to 4; pad col is 0 in both buffers) ----
#pragma unroll
  for (int i = 0; i < 2; ++i) { va[i] = v8zero(); ta[i] = v8zero(); }
  dual_gemm<1, 2>(valbuf, tanbuf, STAGE_STRIDE, smem + OFF_D1F, lane, va, ta);
  activate_stage<2>(va, ta, smem + OFF_C1, valbuf, tanbuf, lane);

  // ---- decoder layer 2: 32 -> 64 ----
#pragma unroll
  for (int i = 0; i < 4; ++i) { va[i] = v8zero(); ta[i] = v8zero(); }
  dual_gemm<8, 4>(valbuf, tanbuf, STAGE_STRIDE, smem + OFF_D2F, lane, va, ta);
  activate_stage<4>(va, ta, smem + OFF_C2, valbuf, tanbuf, lane);

  // ---- decoder layer 3 (linear): 64 -> 128, write xb/dxb straight to HBM ----
#pragma unroll
  for (int i = 0; i < 8; ++i) { va[i] = v8zero(); ta[i] = v8zero(); }
  dual_gemm<16, 8>(valbuf, tanbuf, STAGE_STRIDE, smem + OFF_D3F, lane, va, ta);
#pragma unroll
  for (int nt = 0; nt < 8; ++nt) {
    int col = nt * 16 + c;
    float bb = smem[OFF_C3 + col];
#pragma unroll
    for (int r = 0; r < 8; ++r) {
      int row = r + half * 8;
      size_t g = (size_t)(row0 + row);
      out_xb [g * 128 + col] = va[nt][r] + bb;
      out_dxb[g * 128 + col] = ta[nt][r];
    }
  }
}

extern "C" void kernel_launch(void* const* d_in, const int* in_sizes, int n_in,
                              void* d_out, int out_size, void* d_ws, size_t ws_size,
                              hipStream_t stream) {
  (void)n_in; (void)out_size; (void)d_ws; (void)ws_size;
  const float* x  = (const float*)d_in[0];
  const float* dx = (const float*)d_in[1];
  const float* W1 = (const float*)d_in[2];
  const float* b1 = (const float*)d_in[3];
  const float* W2 = (const float*)d_in[4];
  const float* b2 = (const float*)d_in[5];
  const float* W3 = (const float*)d_in[6];
  const float* b3 = (const float*)d_in[7];
  const float* V1 = (const float*)d_in[8];
  const float* c1 = (const float*)d_in[9];
  const float* V2 = (const float*)d_in[10];
  const float* c2 = (const float*)d_in[11];
  const float* V3 = (const float*)d_in[12];
  const float* c3 = (const float*)d_in[13];
  const float* Ew = (const float*)d_in[14];
  const float* Eb = (const float*)d_in[15];
  float* out = (float*)d_out;

  int n = in_sizes[0] / 128;
  int tiles = (n + 15) / 16;
  int blocks = (tiles + WAVES_PER_BLOCK - 1) / WAVES_PER_BLOCK;
  size_t shmem = (size_t)SMEM_FLOATS * sizeof(float);

  hipLaunchKernelGGL(sindy_ae_kernel, dim3(blocks), dim3(32 * WAVES_PER_BLOCK), shmem, stream,
                     x, dx, W1, b1, W2, b2, W3, b3, V1, c1, V2, c2, V3, c3, Ew, Eb, out, n);
}